// DenseKANLayer_38405597561239
// MI455X (gfx1250) — compile-verified
//
#include <hip/hip_runtime.h>
#include <hip/hip_bf16.h>

typedef float v2f __attribute__((ext_vector_type(2)));
typedef float v8f __attribute__((ext_vector_type(8)));

#define N_IN   128
#define N_OUT  128
#define BATCH  512
#define NB     11            // G + K = 8 + 3 basis functions per input
#define KPI    12            // 11 basis + 1 silu slot per input
#define KDIM   (N_IN * KPI)  // 1536 contraction length

// ---------------------------------------------------------------------------
// Phase 1a: build A^T (K-major, K-pair interleaved).
//   For kk = i*12+g, pair p = kk/2:  At[(p*BATCH + b)*2 + (kk&1)] = value
//   value = B_g(x[b,i]) for g<11, silu(x[b,i]) for g==11.
// One thread per (i,b); consecutive threads share i, vary b -> coalesced writes.
// ---------------------------------------------------------------------------
__global__ void kan_build_A(const float* __restrict__ x,
                            const float* __restrict__ knots,
                            float* __restrict__ At) {
    int idx = blockIdx.x * blockDim.x + threadIdx.x;   // idx = i*BATCH + b
    int b = idx & (BATCH - 1);
    int i = idx >> 9;

    float xv = x[b * N_IN + i];

    // Knot rows are identical across edges; read row 0 (hits scalar/L2 cache).
    float t[2 * 3 + 8 + 1];  // 15 knots
#pragma unroll
    for (int j = 0; j < 15; ++j) t[j] = knots[j];

    // Order-0 indicator, then de Boor recursion exactly as the reference.
    float bb[14];
#pragma unroll
    for (int j = 0; j < 14; ++j)
        bb[j] = (xv >= t[j] && xv < t[j + 1]) ? 1.0f : 0.0f;
#pragma unroll
    for (int order = 1; order <= 3; ++order) {
#pragma unroll
        for (int j = 0; j < 14 - order; ++j) {
            float left  = (xv - t[j]) / (t[j + order] - t[j]);
            float right = (t[j + order + 1] - xv) / (t[j + order + 1] - t[j + 1]);
            bb[j] = left * bb[j] + right * bb[j + 1];
        }
    }

    int base = i * KPI;  // even
#pragma unroll
    for (int g = 0; g < NB; ++g) {
        int kk = base + g;
        At[(kk >> 1) * (BATCH * 2) + b * 2 + (kk & 1)] = bb[g];
    }
    float s = xv / (1.0f + expf(-xv));        // silu = x * sigmoid(x)
    int kk = base + NB;                        // kk&1 == 1
    At[(kk >> 1) * (BATCH * 2) + b * 2 + 1] = s;
}

// ---------------------------------------------------------------------------
// Phase 1b: build W^T (K-major, K-pair interleaved), folding c_spl/c_res.
//   Wt[(p*N_OUT + n)*2 + (kk&1)] : kk=i*12+g -> c_spl[n,i]*c_basis[n*128+i,g],
//                                  kk=i*12+11 -> c_res[n,i]
// ---------------------------------------------------------------------------
__global__ void kan_build_W(const float* __restrict__ c_basis,
                            const float* __restrict__ c_spl,
                            const float* __restrict__ c_res,
                            float* __restrict__ Wt) {
    int idx = blockIdx.x * blockDim.x + threadIdx.x;   // idx = i*N_OUT + n
    int n = idx & (N_OUT - 1);
    int i = idx >> 7;

    int e = n * N_IN + i;                // edge index o*128 + i
    float cs = c_spl[e];
    float cr = c_res[e];

    int base = i * KPI;
#pragma unroll
    for (int g = 0; g < NB; ++g) {
        int kk = base + g;
        Wt[(kk >> 1) * (N_OUT * 2) + n * 2 + (kk & 1)] = cs * c_basis[e * NB + g];
    }
    Wt[((base + NB) >> 1) * (N_OUT * 2) + n * 2 + 1] = cr;
}

// ---------------------------------------------------------------------------
// Phase 2: out(512x128) = A'(512x1536) * W'^T(1536x128) + bias
// One wave per 16x16 output tile, V_WMMA_F32_16X16X4_F32, 384 K-steps.
// A-frag layout (ISA 16x4 f32): lane = half*16 + r16, M=r16;
//   VGPR0 = K = k + 2*half, VGPR1 = K+1  -> one float2 load from K-pair layout.
// B-frag mirrors with N = r16.
// C layout: VGPR r -> M = r + 8*half, N = r16.
// ---------------------------------------------------------------------------
__global__ void kan_gemm_wmma(const float* __restrict__ At,
                              const float* __restrict__ Wt,
                              const float* __restrict__ bias,
                              float* __restrict__ out) {
    int wave = (blockIdx.x * blockDim.x + threadIdx.x) >> 5;  // 0..255
    int lane = threadIdx.x & 31;
    int tm = wave >> 3;     // 0..31 batch tiles
    int tn = wave & 7;      // 0..7  output tiles
    int half = lane >> 4;   // selects K+0/1 vs K+2/3
    int r16  = lane & 15;
    int m = tm * 16 + r16;  // batch row for A-frag
    int n = tn * 16 + r16;  // out col for B-frag / store

    v8f c = {};
#pragma unroll 4
    for (int p = 0; p < KDIM / 2; p += 2) {   // p = K-pair index, step = K of 4
        int pp = p + half;
        v2f a  = *(const v2f*)(At + (size_t)(pp * BATCH + m) * 2);
        v2f bf = *(const v2f*)(Wt + (size_t)(pp * N_OUT + n) * 2);
        c = __builtin_amdgcn_wmma_f32_16x16x4_f32(
                /*neg_a=*/false, a, /*neg_b=*/false, bf,
                /*c_mod=*/(short)0, c, /*reuse_a=*/false, /*reuse_b=*/false);
    }

    float bv = bias[n];
#pragma unroll
    for (int r = 0; r < 8; ++r) {
        int row = tm * 16 + r + 8 * half;
        out[row * N_OUT + n] = c[r] + bv;
    }
}

// ---------------------------------------------------------------------------
extern "C" void kernel_launch(void* const* d_in, const int* in_sizes, int n_in,
                              void* d_out, int out_size, void* d_ws, size_t ws_size,
                              hipStream_t stream) {
    const float* x       = (const float*)d_in[0];  // (512,128)
    const float* knots   = (const float*)d_in[1];  // (16384,15), rows identical
    const float* c_basis = (const float*)d_in[2];  // (16384,11)
    const float* c_spl   = (const float*)d_in[3];  // (128,128)
    const float* c_res   = (const float*)d_in[4];  // (128,128)
    const float* bias    = (const float*)d_in[5];  // (128,)
    float* out = (float*)d_out;                    // (512,128)

    float* Wt = (float*)d_ws;                                          // 768 KB
    float* At = (float*)((char*)d_ws + (size_t)KDIM * N_OUT * sizeof(float)); // 3 MB

    kan_build_W<<<(N_IN * N_OUT) / 256, 256, 0, stream>>>(c_basis, c_spl, c_res, Wt);
    kan_build_A<<<(N_IN * BATCH) / 256, 256, 0, stream>>>(x, knots, At);
    kan_gemm_wmma<<<64, 128, 0, stream>>>(At, Wt, bias, out);
}